// GCONV_23364622090643
// MI455X (gfx1250) — compile-verified
//
#include <hip/hip_runtime.h>

// Problem constants (match reference)
#define N_NODES 10000
#define BATCH   32
#define IN_F    66
#define OUT_F   64
#define FEAT    (IN_F * BATCH)      // 2112 floats per node row
#define NUM_MAT 5
#define IPAD    68                  // IN_F padded to multiple of 4

typedef __attribute__((ext_vector_type(2))) float v2f;
typedef __attribute__((ext_vector_type(8))) float v8f;

// ---------------------------------------------------------------------------
// x0[n, i*32+b] = inputs[b, n*IN + i]   (write-coalesced transpose)
// ---------------------------------------------------------------------------
__global__ __launch_bounds__(256) void build_x0_kernel(const float* __restrict__ inp,
                                                       float* __restrict__ x0) {
    int idx = blockIdx.x * 256 + threadIdx.x;          // < N*FEAT = 21,120,000
    if (idx >= N_NODES * FEAT) return;
    int n = idx / FEAT;
    int f = idx - n * FEAT;
    int i = f >> 5;                                    // / BATCH (=32)
    int b = f & 31;
    x0[idx] = inp[(size_t)b * (N_NODES * IN_F) + (size_t)n * IN_F + i];
}

// ---------------------------------------------------------------------------
// y[rows[e], :] += vals[e] * x[cols[e], :]    one wave per edge, 66 floats/lane
// x rows are 8.4KB contiguous -> wave-coalesced 128B accesses; states are
// L2-resident (84.5MB each vs 192MB L2), so atomics and gathers stay on-chip.
// ---------------------------------------------------------------------------
__global__ __launch_bounds__(256) void spmm_kernel(const int* __restrict__ rows,
                                                   const int* __restrict__ cols,
                                                   const float* __restrict__ vals,
                                                   const float* __restrict__ x,
                                                   float* __restrict__ y, int nnz) {
    int wave = (blockIdx.x * 256 + threadIdx.x) >> 5;
    int lane = threadIdx.x & 31;
    if (wave >= nnz) return;
    int r = rows[wave];
    int c = cols[wave];
    float v = vals[wave];
    const float* __restrict__ xr = x + (size_t)c * FEAT;
    float* __restrict__ yr = y + (size_t)r * FEAT;
#pragma unroll 6
    for (int j = 0; j < FEAT / 32; ++j) {
        int f = lane + j * 32;
        atomicAdd(&yr[f], v * xr[f]);                  // global_atomic_add_f32
    }
}

// ---------------------------------------------------------------------------
// Fused gather-GEMM with fp32 WMMA, K iterated matrix-major:
//   out[r, o] = sum_m sum_i xs[m][n, i*32+b] * W[i*5+m, o] + bias[o]
//   (xs2 = 2*y2 - x0 and xs4 = 2*y4 - y1 computed on the fly: CHEB=true)
// One wave owns a 16x64 output tile; 8 waves/block; per-m weight rows staged
// in LDS (68 x stride-72, zero-padded) -> guard-free inner loop, linear A gather.
// ---------------------------------------------------------------------------
#define LDSW 72   // 2*72 mod 64 == 16: the two K-rows one ds access touches
                  // (lanes 0-15 row R, lanes 16-31 row R+2) hit disjoint banks

template<bool CHEB>
__device__ __forceinline__ void process_mat(const float* __restrict__ pm,
                                            const float* __restrict__ qm,
                                            int m,
                                            const float* __restrict__ W,
                                            float* wlds, int tid, int hi, int mrow,
                                            v8f (&acc)[4]) {
    __syncthreads();                                   // protect previous LDS use
    // stage W[i*5+m, 0..63] for i=0..67 into LDS rows (straight-line, clamped)
    for (int t = tid; t < IPAD * 64; t += 256) {       // 17 iterations exactly
        int i = t >> 6;
        int c = t & 63;
        int isrc = (i < IN_F) ? i : (IN_F - 1);        // clamped in-bounds load
        float w = W[isrc * (NUM_MAT * OUT_F) + m * OUT_F + c];
        wlds[i * LDSW + c] = (i < IN_F) ? w : 0.0f;    // cndmask zero-pad
    }
    __syncthreads();

    // steps 0..15 cover i = 0..63: no guards anywhere
#pragma unroll 4
    for (int ks = 0; ks < 16; ++ks) {
        const int i0 = ks * 4 + 2 * hi;
        float a0 = pm[i0 * 32];
        float a1 = pm[(i0 + 1) * 32];
        if (CHEB) {
            a0 = 2.0f * a0 - qm[i0 * 32];
            a1 = 2.0f * a1 - qm[(i0 + 1) * 32];
        }
        v2f afrag;
        afrag.x = a0;
        afrag.y = a1;
        const int krl = ks * 4 + 2 * hi;
#pragma unroll
        for (int t = 0; t < 4; ++t) {
            v2f bfrag;                                 // B(4x16) mirrors A layout
            bfrag.x = wlds[krl * LDSW + t * 16 + mrow];
            bfrag.y = wlds[(krl + 1) * LDSW + t * 16 + mrow];
            acc[t] = __builtin_amdgcn_wmma_f32_16x16x4_f32(
                false, afrag, false, bfrag, (short)0, acc[t], false, false);
        }
    }
    // epilogue step: i = 64..67 (valid A data only for hi==0 lanes: i=64,65)
    {
        float a0 = 0.0f, a1 = 0.0f;
        if (hi == 0) {
            a0 = pm[64 * 32];
            a1 = pm[65 * 32];
            if (CHEB) {
                a0 = 2.0f * a0 - qm[64 * 32];
                a1 = 2.0f * a1 - qm[65 * 32];
            }
        }
        v2f afrag;
        afrag.x = a0;
        afrag.y = a1;
        const int krl = 64 + 2 * hi;                   // LDS rows 66,67 are zero
#pragma unroll
        for (int t = 0; t < 4; ++t) {
            v2f bfrag;
            bfrag.x = wlds[krl * LDSW + t * 16 + mrow];
            bfrag.y = wlds[(krl + 1) * LDSW + t * 16 + mrow];
            acc[t] = __builtin_amdgcn_wmma_f32_16x16x4_f32(
                false, afrag, false, bfrag, (short)0, acc[t], false, false);
        }
    }
}

__global__ __launch_bounds__(256) void gemm_wmma_kernel(const float* __restrict__ x0,
                                                        const float* __restrict__ y1,
                                                        const float* __restrict__ y2,
                                                        const float* __restrict__ y3,
                                                        const float* __restrict__ y4,
                                                        const float* __restrict__ W,
                                                        const float* __restrict__ bias,
                                                        float* __restrict__ out) {
    __shared__ float wlds[IPAD * LDSW];                // 19,584 B

    const int tid    = threadIdx.x;
    const int lane   = tid & 31;
    const int waveId = tid >> 5;
    const int mrow   = lane & 15;                      // row (A) / col (B,C,D)
    const int hi     = lane >> 4;                      // K-pair half / M+8 half

    const int rt = blockIdx.x * 8 + waveId;            // 16-row tile, [0, 20000)
    const int r0 = rt * 16;                            // N%16==0 -> constant b
    const int b  = r0 / N_NODES;
    const int n  = (r0 - b * N_NODES) + mrow;
    const size_t gofs = (size_t)n * FEAT + b;

    const float* p0 = x0 + gofs;
    const float* p1 = y1 + gofs;
    const float* p2 = y2 + gofs;
    const float* p3 = y3 + gofs;
    const float* p4 = y4 + gofs;

    // C/D layout: VGPR j -> row r0 + j + 8*hi, col mrow + 16*t; bias folded in
    v8f acc[4];
#pragma unroll
    for (int t = 0; t < 4; ++t) {
        float bv = bias[t * 16 + mrow];
#pragma unroll
        for (int j = 0; j < 8; ++j) acc[t][j] = bv;
    }

    process_mat<false>(p0, p0, 0, W, wlds, tid, hi, mrow, acc); // x0
    process_mat<false>(p1, p1, 1, W, wlds, tid, hi, mrow, acc); // A1 x0
    process_mat<true >(p2, p0, 2, W, wlds, tid, hi, mrow, acc); // 2*y2 - x0
    process_mat<false>(p3, p3, 3, W, wlds, tid, hi, mrow, acc); // A2 y1
    process_mat<true >(p4, p1, 4, W, wlds, tid, hi, mrow, acc); // 2*y4 - y1

    // store D: row = r0 + j + 8*hi, col = t*16 + mrow   (out = [B*N, 64] row-major)
#pragma unroll
    for (int t = 0; t < 4; ++t)
#pragma unroll
        for (int j = 0; j < 8; ++j)
            out[(size_t)(r0 + j + 8 * hi) * OUT_F + t * 16 + mrow] = acc[t][j];
}

// ---------------------------------------------------------------------------
// Launch: needs ws >= 5 * N * FEAT * 4 B = 422.4 MB scratch.
// ---------------------------------------------------------------------------
extern "C" void kernel_launch(void* const* d_in, const int* in_sizes, int n_in,
                              void* d_out, int out_size, void* d_ws, size_t ws_size,
                              hipStream_t stream) {
    const float* inputs = (const float*)d_in[0];
    const float* vals1  = (const float*)d_in[1];
    const int*   rows1  = (const int*)d_in[2];
    const int*   cols1  = (const int*)d_in[3];
    const float* vals2  = (const float*)d_in[4];
    const int*   rows2  = (const int*)d_in[5];
    const int*   cols2  = (const int*)d_in[6];
    const float* W      = (const float*)d_in[7];
    const float* bias   = (const float*)d_in[8];
    float*       out    = (float*)d_out;
    const int    nnz    = in_sizes[1];

    const size_t MATSZ = (size_t)N_NODES * FEAT;       // 21,120,000 floats
    float* x0 = (float*)d_ws;
    float* y1 = x0 + MATSZ;
    float* y2 = y1 + MATSZ;
    float* y3 = y2 + MATSZ;
    float* y4 = y3 + MATSZ;

    // zero the four accumulation targets (graph-capturable)
    hipMemsetAsync(y1, 0, 4 * MATSZ * sizeof(float), stream);

    const int egrid = (int)((MATSZ + 255) / 256);
    build_x0_kernel<<<egrid, 256, 0, stream>>>(inputs, x0);

    const int sgrid = (nnz + 7) / 8;                   // 8 edge-waves per block
    spmm_kernel<<<sgrid, 256, 0, stream>>>(rows1, cols1, vals1, x0, y1, nnz);
    spmm_kernel<<<sgrid, 256, 0, stream>>>(rows1, cols1, vals1, y1, y2, nnz);
    spmm_kernel<<<sgrid, 256, 0, stream>>>(rows2, cols2, vals2, y1, y3, nnz);
    spmm_kernel<<<sgrid, 256, 0, stream>>>(rows2, cols2, vals2, y3, y4, nnz);

    const int ggrid = (BATCH * N_NODES) / (16 * 8);    // 2500 blocks, 8 waves each
    gemm_wmma_kernel<<<ggrid, 256, 0, stream>>>(x0, y1, y2, y3, y4, W, bias, out);
}